// PointNetLossForSemanticSegmentation_18279380812443
// MI455X (gfx1250) — compile-verified
//
#include <hip/hip_runtime.h>

typedef float v2f __attribute__((ext_vector_type(2)));
typedef float v8f __attribute__((ext_vector_type(8)));

#define BN_POINTS (32 * 16384)   // B*N = 524288
#define NCLS 50
#define BLOCK 256
#define WAVES (BLOCK / 32)

// ---------------------------------------------------------------------------
// Init: zero the 3 global accumulators in workspace (graph-capture safe).
// ---------------------------------------------------------------------------
__global__ void pnl_init(float* __restrict__ gacc) {
    if (threadIdx.x < 3) gacc[threadIdx.x] = 0.0f;
}

// ---------------------------------------------------------------------------
// Main: per-point log-softmax stats + match count, block-reduced.
// Cross-wave reduction is performed on the matrix pipe via
// v_wmma_f32_16x16x4_f32 (A = ones, per-wave partials along K of B).
// ---------------------------------------------------------------------------
__global__ __launch_bounds__(BLOCK) void pnl_main(
    const float* __restrict__ pred,        // [BN_POINTS, 50]
    const int*   __restrict__ targets,     // [BN_POINTS]
    const int*   __restrict__ pred_choice, // [BN_POINTS]
    float* __restrict__ gacc)              // [3]: sum(-logp_t), sum(1-pn), matches
{
    __shared__ float wsum[WAVES][3];

    const int p = blockIdx.x * BLOCK + threadIdx.x;

    float acc_nll = 0.0f;   // -log p(target)
    float acc_foc = 0.0f;   // 1 - p(target)
    float acc_mat = 0.0f;   // pred_choice == target

    if (p < BN_POINTS) {
        const float* pp  = pred + (size_t)p * NCLS;
        const v2f*   pp2 = reinterpret_cast<const v2f*>(pp);  // 8B aligned (stride 200B)

        v2f v[NCLS / 2];
        #pragma unroll
        for (int i = 0; i < NCLS / 2; ++i) v[i] = pp2[i];

        float m = -3.402823466e38f;
        #pragma unroll
        for (int i = 0; i < NCLS / 2; ++i) m = fmaxf(m, fmaxf(v[i].x, v[i].y));

        float s = 0.0f;
        #pragma unroll
        for (int i = 0; i < NCLS / 2; ++i)
            s += __expf(v[i].x - m) + __expf(v[i].y - m);

        const int t  = targets[p];
        const float xt = pp[t];                 // cached re-read, avoids VGPR indexing
        const float logp_t = xt - m - __logf(s);
        const float pn = __expf(logp_t);

        acc_nll = -logp_t;
        acc_foc = 1.0f - pn;
        acc_mat = (pred_choice[p] == t) ? 1.0f : 0.0f;
    }

    // wave32 tree reduction
    #pragma unroll
    for (int off = 16; off >= 1; off >>= 1) {
        acc_nll += __shfl_down(acc_nll, off);
        acc_foc += __shfl_down(acc_foc, off);
        acc_mat += __shfl_down(acc_mat, off);
    }

    const int lane = threadIdx.x & 31;
    const int wave = threadIdx.x >> 5;
    if (lane == 0) {
        wsum[wave][0] = acc_nll;
        wsum[wave][1] = acc_foc;
        wsum[wave][2] = acc_mat;
    }
    __syncthreads();

    // Wave 0 (full EXEC, as WMMA requires) sums the 8 per-wave partials with
    // one V_WMMA_F32_16X16X4_F32:
    //   B(4x16): col n<3      holds waves 0..3 of accumulator n along K
    //            col 8<=n<11  holds waves 4..7 of accumulator n-8 along K
    //   A = ones  =>  D[0][n] = sum_k B[k][n]
    if (threadIdx.x < 32) {
        const int L  = threadIdx.x;
        const int n  = L & 15;
        const int k0 = (L < 16) ? 0 : 1;   // B vgpr0 row: K=0 (lanes 0-15), K=1 (16-31)
        const int k1 = k0 + 2;             // B vgpr1 row: K=2 / K=3

        float b0 = 0.0f, b1 = 0.0f;
        if (n < 3)                    { b0 = wsum[k0][n];         b1 = wsum[k1][n]; }
        else if (n >= 8 && n < 11)    { b0 = wsum[k0 + 4][n - 8]; b1 = wsum[k1 + 4][n - 8]; }

        v2f A = {1.0f, 1.0f};
        v2f B = {b0, b1};
        v8f Cm = {};
        v8f D = __builtin_amdgcn_wmma_f32_16x16x4_f32(
            /*neg_a=*/false, A, /*neg_b=*/false, B,
            /*c_mod=*/(short)0, Cm, /*reuse_a=*/false, /*reuse_b=*/false);

        float r = D[0];                 // lane n: column-sum over K (waves group)
        r += __shfl_down(r, 8);         // fold waves 4..7 group into lanes 0..2
        if (L < 3) atomicAdd(&gacc[L], r);
    }
}

// ---------------------------------------------------------------------------
// Finalize: combine the three sums into the scalar loss.
// ---------------------------------------------------------------------------
__global__ void pnl_finalize(const float* __restrict__ gacc, float* __restrict__ out) {
    const float M = (float)BN_POINTS;
    const float a = gacc[0];            // sum(-logp_t)
    const float b = gacc[1];            // sum(1 - pn)
    const float c = gacc[2];            // match count
    const float ce    = a / M;
    const float loss  = (b / M) * ce;   // gamma == 1
    const float dice  = 1.0f - 2.0f * (c + 1.0f) / (2.0f * M + 1.0f);
    out[0] = loss + dice;
}

extern "C" void kernel_launch(void* const* d_in, const int* in_sizes, int n_in,
                              void* d_out, int out_size, void* d_ws, size_t ws_size,
                              hipStream_t stream) {
    (void)in_sizes; (void)n_in; (void)out_size; (void)ws_size;
    const float* pred        = (const float*)d_in[0];
    const int*   targets     = (const int*)d_in[1];
    const int*   pred_choice = (const int*)d_in[2];
    float* gacc = (float*)d_ws;
    float* out  = (float*)d_out;

    pnl_init<<<1, 32, 0, stream>>>(gacc);
    pnl_main<<<BN_POINTS / BLOCK, BLOCK, 0, stream>>>(pred, targets, pred_choice, gacc);
    pnl_finalize<<<1, 1, 0, stream>>>(gacc, out);
}